// QuantumNeuralNetwork_50964081934731
// MI455X (gfx1250) — compile-verified
//
#include <hip/hip_runtime.h>
#include <math.h>

// QuantumNeuralNetwork — MI455X (gfx1250) implementation.
// Memory-bound streaming reduction (160 MB @ 23.3 TB/s ~ 7us) + tiny WMMA GEMM.

#define NQ     20
#define NL     8
#define BATCH  32
#define DIM    (1 << NQ)          // 1048576
#define NCHUNK 64
#define CHUNK  (DIM / NCHUNK)     // 16384 elements per (b, chunk) block
#define K2T    256                // threads per block in reduce kernel
#define K2IT   (CHUNK / (K2T * 4))// 16 iterations of float4 per thread

typedef __attribute__((ext_vector_type(2))) float v2f;
typedef __attribute__((ext_vector_type(8))) float v8f;

// ---------------------------------------------------------------------------
// Kernel 1: wsq[i] = (prod_l noise[l][i])^2   (had_factor^8 cancels: even sign
// power + renorm kills the 2^-80 constant). 36 MB of traffic, float4 streams.
// ---------------------------------------------------------------------------
__global__ void qnn_noise_sq(const float* __restrict__ noise,
                             float* __restrict__ wsq) {
  const int i = (blockIdx.x * blockDim.x + threadIdx.x) * 4;
  float4 p = *(const float4*)(noise + i);
#pragma unroll
  for (int l = 1; l < NL; ++l) {
    float4 n = *(const float4*)(noise + (size_t)l * DIM + i);
    p.x *= n.x; p.y *= n.y; p.z *= n.z; p.w *= n.w;
  }
  float4 o;
  o.x = p.x * p.x; o.y = p.y * p.y; o.z = p.z * p.z; o.w = p.w * p.w;
  *(float4*)(wsq + i) = o;
}

__device__ __forceinline__ float wave_sum32(float v) {
#pragma unroll
  for (int off = 16; off > 0; off >>= 1) v += __shfl_xor(v, off, 32);
  return v;
}

// ---------------------------------------------------------------------------
// Kernel 2: per (batch b, chunk) partial sums of p = amps0^2 * wsq.
// Element index e = chunk*16384 + it*1024 + tid*4 + c:
//   bits 0..1  <- c        (per-float4-component accumulators a0, a1)
//   bits 2..6  <- lane     (masked lane-sums at reduction time)
//   bits 7..9  <- wave id  (uniform per wave; applied at reduction time)
//   bits 10..19<- hi=e>>10 (block-uniform per iteration -> 10 cndmask-adds)
// Deterministic: butterfly shuffles + fixed-order LDS combine, no atomics.
// ---------------------------------------------------------------------------
__global__ void qnn_reduce(const float* __restrict__ amps0,
                           const float* __restrict__ wsq,
                           float* __restrict__ partials) {
  const int tid   = threadIdx.x;
  const int chunk = blockIdx.x;
  const int b     = blockIdx.y;
  const float* __restrict__ arow = amps0 + (size_t)b * DIM + (size_t)chunk * CHUNK;
  const float* __restrict__ wrow = wsq + (size_t)chunk * CHUNK;

  float tot = 0.f, a0 = 0.f, a1 = 0.f;
  float h[10];
#pragma unroll
  for (int j = 0; j < 10; ++j) h[j] = 0.f;

#pragma unroll 4
  for (int it = 0; it < K2IT; ++it) {
    const int off = it * (K2T * 4) + tid * 4;
    float4 a = *(const float4*)(arow + off);
    float4 w = *(const float4*)(wrow + off);
    if (it + 1 < K2IT) {  // gfx1250 global_prefetch_b8
      __builtin_prefetch(arow + off + K2T * 4, 0, 0);
      __builtin_prefetch(wrow + off + K2T * 4, 0, 0);
    }
    const float p0 = a.x * a.x * w.x;
    const float p1 = a.y * a.y * w.y;
    const float p2 = a.z * a.z * w.z;
    const float p3 = a.w * a.w * w.w;
    const float s4 = (p0 + p1) + (p2 + p3);
    tot += s4;
    a0  += p1 + p3;          // element bit 0 set (c = 1,3)
    a1  += p2 + p3;          // element bit 1 set (c = 2,3)
    const unsigned hi = (unsigned)(chunk * (CHUNK >> 10) + it);  // uniform
#pragma unroll
    for (int j = 0; j < 10; ++j)
      h[j] += ((hi >> j) & 1u) ? s4 : 0.f;   // element bits 10..19
  }

  const int lane = tid & 31;
  const int wv   = tid >> 5;

  float m[5];
#pragma unroll
  for (int k = 0; k < 5; ++k)              // element bits 2..6 <- lane bits
    m[k] = wave_sum32(((lane >> k) & 1) ? tot : 0.f);
  const float ts = wave_sum32(tot);
  const float s0 = wave_sum32(a0);
  const float s1 = wave_sum32(a1);
  float hs[10];
#pragma unroll
  for (int j = 0; j < 10; ++j) hs[j] = wave_sum32(h[j]);

  __shared__ float part[8][21];
  if (lane == 0) {
    part[wv][0] = s0;
    part[wv][1] = s1;
#pragma unroll
    for (int k = 0; k < 5; ++k) part[wv][2 + k] = m[k];
    part[wv][7] = (wv & 1) ? ts : 0.f;     // element bits 7..9 <- wave id
    part[wv][8] = (wv & 2) ? ts : 0.f;
    part[wv][9] = (wv & 4) ? ts : 0.f;
#pragma unroll
    for (int j = 0; j < 10; ++j) part[wv][10 + j] = hs[j];
    part[wv][20] = ts;                     // grand total
  }
  __syncthreads();
  if (tid < 21) {
    float s = 0.f;
#pragma unroll
    for (int w = 0; w < 8; ++w) s += part[w][tid];
    partials[((size_t)(b * NCHUNK + chunk)) * 32 + tid] = s;
  }
}

// ---------------------------------------------------------------------------
// Kernel 3: single wave (EXEC all-1s, as WMMA requires).
// Sum chunk partials -> qubit_probs[32][20] -> tanh(qp @ Wi + bi) via
// V_WMMA_F32_16X16X4_F32: 2x2 output tiles of 16x16, 5 chained K=4 steps.
// ---------------------------------------------------------------------------
__global__ void qnn_final(const float* __restrict__ partials,
                          const float* __restrict__ Wi,
                          const float* __restrict__ bi,
                          float* __restrict__ out) {
  __shared__ float Qs[BATCH * 21];
  __shared__ float qp[BATCH * NQ];
  const int lane = threadIdx.x;

  for (int idx = lane; idx < BATCH * 21; idx += 32) {
    const int b = idx / 21, s = idx % 21;
    float acc = 0.f;
    for (int c = 0; c < NCHUNK; ++c)
      acc += partials[((size_t)(b * NCHUNK + c)) * 32 + s];
    Qs[idx] = acc;
  }
  __syncthreads();
  for (int idx = lane; idx < BATCH * NQ; idx += 32) {
    const int b = idx / NQ, q = idx % NQ;
    qp[idx] = Qs[b * 21 + q] / Qs[b * 21 + 20];
  }
  __syncthreads();

  const int nlo  = lane & 15;   // A: M=lane&15 ; B/C/D: N=lane&15
  const int half = lane >> 4;   // A/B: K offset +2 ; C/D: M offset +8

#pragma unroll
  for (int mt = 0; mt < 2; ++mt) {
#pragma unroll
    for (int nt = 0; nt < 2; ++nt) {
      const int n = nt * 16 + nlo;
      const float bias = (n < NQ) ? bi[n] : 0.f;   // bias is per-N, same all M
      v8f c;
#pragma unroll
      for (int r = 0; r < 8; ++r) c[r] = bias;
#pragma unroll
      for (int ks = 0; ks < 5; ++ks) {             // K = 20 = 5 x 4
        v2f A, Bv;
#pragma unroll
        for (int j = 0; j < 2; ++j) {
          const int k = ks * 4 + j + 2 * half;     // k in [0,20) always
          A[j]  = qp[(mt * 16 + nlo) * NQ + k];    // A 16x4: M=nlo
          Bv[j] = (n < NQ) ? Wi[k * NQ + n] : 0.f; // B 4x16: K rows
        }
        c = __builtin_amdgcn_wmma_f32_16x16x4_f32(
            false, A, false, Bv, (short)0, c, false, false);
      }
#pragma unroll
      for (int r = 0; r < 8; ++r) {                // C/D: VGPR r -> M=r(+8)
        const int M = mt * 16 + r + half * 8;
        if (n < NQ) out[M * NQ + n] = tanhf(c[r]);
      }
    }
  }
}

// ---------------------------------------------------------------------------
extern "C" void kernel_launch(void* const* d_in, const int* in_sizes, int n_in,
                              void* d_out, int out_size, void* d_ws, size_t ws_size,
                              hipStream_t stream) {
  // setup_inputs order: 0:x 1:W1 2:b1 3:W2 4:b2 5:W3 6:b3 7:amps0 8:cnot_noise 9:Wi 10:bi
  const float* amps0 = (const float*)d_in[7];
  const float* noise = (const float*)d_in[8];
  const float* Wi    = (const float*)d_in[9];
  const float* bi    = (const float*)d_in[10];
  float* out = (float*)d_out;

  float* wsq      = (float*)d_ws;       // DIM floats (4 MB)
  float* partials = wsq + DIM;          // BATCH*NCHUNK*32 floats (256 KB)

  qnn_noise_sq<<<DIM / (256 * 4), 256, 0, stream>>>(noise, wsq);
  qnn_reduce<<<dim3(NCHUNK, BATCH), K2T, 0, stream>>>(amps0, wsq, partials);
  qnn_final<<<1, 32, 0, stream>>>(partials, Wi, bi, out);
}